// NodeEmbeddingLayer_16870631538737
// MI455X (gfx1250) — compile-verified
//
#include <hip/hip_runtime.h>

// ---------------------------------------------------------------------------
// Problem constants (from the reference)
// ---------------------------------------------------------------------------
#define FDIM 256   // node feature dim
#define EXP  16    // experts (node types)
#define DDIM 256   // embedding dim
#define HDIM 512   // hidden dim
#define GG   256   // graphs (selected nodes)
#define PADR 16    // zero-pad rows appended to xg so unguarded A reads stay in-bounds

// f32 WMMA fragment types (16x16x4: A/B = 2 VGPRs, C/D = 8 VGPRs per lane)
typedef __attribute__((ext_vector_type(2))) float v2f;
typedef __attribute__((ext_vector_type(8))) float v8f;

// ---------------------------------------------------------------------------
// CDNA5 async global->LDS copy (ASYNCcnt path), with safe fallback.
// Builtin signature (from hipcc diagnostic): params are pointers to
// 'int __attribute__((vector_size(16)))' in addrspace(1) / addrspace(3).
// ---------------------------------------------------------------------------
#if defined(__has_builtin)
#if __has_builtin(__builtin_amdgcn_global_load_async_to_lds_b128) && \
    __has_builtin(__builtin_amdgcn_s_wait_asynccnt)
#define HAVE_ASYNC_LDS 1
#endif
#endif

typedef int v4i __attribute__((vector_size(16)));
#ifdef HAVE_ASYNC_LDS
using gv4i = __attribute__((address_space(1))) v4i;
using lv4i = __attribute__((address_space(3))) v4i;
#endif

// Copy nfloats (multiple of 128) contiguous floats global->LDS with one wave.
__device__ __forceinline__ void wave_copy_to_lds(const float* gsrc, float* ldst,
                                                 int lane, int nfloats) {
#ifdef HAVE_ASYNC_LDS
    for (int c = lane; c < (nfloats >> 2); c += 32) {
        __builtin_amdgcn_global_load_async_to_lds_b128(
            (gv4i*)(uintptr_t)(gsrc + c * 4),
            (lv4i*)(unsigned int)(uintptr_t)(ldst + c * 4),
            0, 0);
    }
    __builtin_amdgcn_s_wait_asynccnt(0);
#else
    for (int c = lane; c < (nfloats >> 2); c += 32) {
        float4 v = ((const float4*)gsrc)[c];
        ((float4*)ldst)[c] = v;
    }
#endif
}

// ---------------------------------------------------------------------------
// Workspace layout (bytes). Total ~1.07 MB.
// ---------------------------------------------------------------------------
#define WS_PERM   0                              // int[GG]      slot -> g
#define WS_SLOTOF 1024                           // int[GG]      g -> slot
#define WS_OFF    2048                           // int[EXP+1]   expert offsets
#define WS_XG     4096                           // float[(GG+PADR)*FDIM]
#define WS_EMB    (WS_XG  + (GG + PADR) * FDIM * 4)  // float[GG*DDIM]
#define WS_H      (WS_EMB + GG * DDIM * 4)           // float[GG*HDIM]

// ---------------------------------------------------------------------------
// Kernel 1: counting-sort the 256 selected rows by expert type.
// ---------------------------------------------------------------------------
__global__ void bucket_kernel(const int* __restrict__ node_type,
                              const int* __restrict__ g_node_id,
                              int* __restrict__ perm, int* __restrict__ slot_of,
                              int* __restrict__ eoff) {
    __shared__ int cnt[EXP];
    __shared__ int base[EXP + 1];
    __shared__ int cur[EXP];
    int g = threadIdx.x;
    if (g < EXP) { cnt[g] = 0; cur[g] = 0; }
    __syncthreads();
    int t = node_type[g_node_id[g]];
    atomicAdd(&cnt[t], 1);
    __syncthreads();
    if (g == 0) {
        int s = 0;
        for (int e = 0; e < EXP; ++e) { base[e] = s; s += cnt[e]; }
        base[EXP] = s;
    }
    __syncthreads();
    int slot = base[t] + atomicAdd(&cur[t], 1);
    perm[slot]  = g;
    slot_of[g]  = slot;
    if (g <= EXP) eoff[g] = base[g];
}

// ---------------------------------------------------------------------------
// Kernel 2: gather + mask selected rows into slot order; zero the pad rows.
// grid = GG + PADR blocks of FDIM threads.
// ---------------------------------------------------------------------------
__global__ void gather_kernel(const float* __restrict__ node_fea,
                              const float* __restrict__ masks,
                              const int* __restrict__ g_node_id,
                              const int* __restrict__ perm,
                              float* __restrict__ xg) {
    int slot = blockIdx.x;
    int f = threadIdx.x;
    if (slot >= GG) {                      // zero pad rows
        xg[slot * FDIM + f] = 0.0f;
        return;
    }
    int n = g_node_id[perm[slot]];
    size_t src = (size_t)n * FDIM + f;
    xg[slot * FDIM + f] = node_fea[src] * masks[src];
}

// ---------------------------------------------------------------------------
// WMMA fragment layouts (ISA 7.12.2, wave32):
//   A 16x4 f32 : M = lane&15; lane<16 -> K={0,1}, lane>=16 -> K={2,3}
//   B 4x16 f32 : N = lane&15; lane<16 -> K={0,1}, lane>=16 -> K={2,3}
//   C 16x16 f32: VGPR r -> M = r + (lane>=16 ? 8 : 0); N = lane&15
// Each wave computes a 16x64 tile: A panel staged in LDS feeds 4 WMMAs/k-step.
// ---------------------------------------------------------------------------

// ---------------------------------------------------------------------------
// Kernel 3: grouped (per-expert) GEMM  emb = xg @ W_emb[e] + b_emb[e]
// grid = (DDIM/64, GG/16, EXP); block = 1 wave (32 lanes)
// ---------------------------------------------------------------------------
__global__ void expert_gemm_kernel(const float* __restrict__ xg,
                                   const float* __restrict__ W_emb,
                                   const float* __restrict__ b_emb,
                                   const int* __restrict__ eoff,
                                   float* __restrict__ emb) {
    __shared__ float As[16 * FDIM];           // 16 KB A panel
    int e = blockIdx.z;
    int off0 = eoff[e], off1 = eoff[e + 1];
    int row0 = off0 + blockIdx.y * 16;
    if (row0 >= off1) return;                 // wave-uniform early exit
    int col0 = blockIdx.x * 64;

    int lane = threadIdx.x;
    int arow = lane & 15;
    int kofs = (lane >> 4) << 1;              // 0 for lanes 0-15, 2 for 16-31
    const float* W = W_emb + (size_t)e * FDIM * DDIM;

    // Stage A panel (rows row0..row0+15; pad rows are zero) into LDS.
    wave_copy_to_lds(xg + (size_t)row0 * FDIM, As, lane, 16 * FDIM);

    v8f c0 = {}, c1 = {}, c2 = {}, c3 = {};
    int ncol = col0 + (lane & 15);
    for (int k = 0; k < FDIM; k += 4) {
        v2f a, b;
        a[0] = As[arow * FDIM + k + kofs];
        a[1] = As[arow * FDIM + k + kofs + 1];
        const float* Wr0 = W + (size_t)(k + kofs) * DDIM + ncol;
        const float* Wr1 = Wr0 + DDIM;
        b[0] = Wr0[0];  b[1] = Wr1[0];
        c0 = __builtin_amdgcn_wmma_f32_16x16x4_f32(false, a, false, b, (short)0, c0, false, false);
        b[0] = Wr0[16]; b[1] = Wr1[16];
        c1 = __builtin_amdgcn_wmma_f32_16x16x4_f32(false, a, false, b, (short)0, c1, false, false);
        b[0] = Wr0[32]; b[1] = Wr1[32];
        c2 = __builtin_amdgcn_wmma_f32_16x16x4_f32(false, a, false, b, (short)0, c2, false, false);
        b[0] = Wr0[48]; b[1] = Wr1[48];
        c3 = __builtin_amdgcn_wmma_f32_16x16x4_f32(false, a, false, b, (short)0, c3, false, false);
    }

    int mbase = row0 + ((lane >> 4) << 3);
#pragma unroll
    for (int t = 0; t < 4; ++t) {
        v8f c = (t == 0) ? c0 : (t == 1) ? c1 : (t == 2) ? c2 : c3;
        int nn = ncol + t * 16;
        float bias = b_emb[e * DDIM + nn];
#pragma unroll
        for (int r = 0; r < 8; ++r) {
            int m = mbase + r;
            if (m < off1) emb[m * DDIM + nn] = c[r] + bias;
        }
    }
}

// ---------------------------------------------------------------------------
// Kernel 4: dense GEMM  h = emb @ W_hid + b_hid   (256x256 @ 256x512)
// grid = (HDIM/64, GG/16); block = 1 wave
// ---------------------------------------------------------------------------
__global__ void hidden_gemm_kernel(const float* __restrict__ emb,
                                   const float* __restrict__ W_hid,
                                   const float* __restrict__ b_hid,
                                   float* __restrict__ h) {
    __shared__ float As[16 * DDIM];           // 16 KB A panel
    int col0 = blockIdx.x * 64;
    int row0 = blockIdx.y * 16;
    int lane = threadIdx.x;
    int arow = lane & 15;
    int kofs = (lane >> 4) << 1;

    wave_copy_to_lds(emb + (size_t)row0 * DDIM, As, lane, 16 * DDIM);

    v8f c0 = {}, c1 = {}, c2 = {}, c3 = {};
    int ncol = col0 + (lane & 15);
    for (int k = 0; k < DDIM; k += 4) {
        v2f a, b;
        a[0] = As[arow * DDIM + k + kofs];
        a[1] = As[arow * DDIM + k + kofs + 1];
        const float* Wr0 = W_hid + (size_t)(k + kofs) * HDIM + ncol;
        const float* Wr1 = Wr0 + HDIM;
        b[0] = Wr0[0];  b[1] = Wr1[0];
        c0 = __builtin_amdgcn_wmma_f32_16x16x4_f32(false, a, false, b, (short)0, c0, false, false);
        b[0] = Wr0[16]; b[1] = Wr1[16];
        c1 = __builtin_amdgcn_wmma_f32_16x16x4_f32(false, a, false, b, (short)0, c1, false, false);
        b[0] = Wr0[32]; b[1] = Wr1[32];
        c2 = __builtin_amdgcn_wmma_f32_16x16x4_f32(false, a, false, b, (short)0, c2, false, false);
        b[0] = Wr0[48]; b[1] = Wr1[48];
        c3 = __builtin_amdgcn_wmma_f32_16x16x4_f32(false, a, false, b, (short)0, c3, false, false);
    }

    int mbase = row0 + ((lane >> 4) << 3);
#pragma unroll
    for (int t = 0; t < 4; ++t) {
        v8f c = (t == 0) ? c0 : (t == 1) ? c1 : (t == 2) ? c2 : c3;
        int nn = ncol + t * 16;
        float bias = b_hid[nn];
#pragma unroll
        for (int r = 0; r < 8; ++r)
            h[(mbase + r) * HDIM + nn] = c[r] + bias;
    }
}

// ---------------------------------------------------------------------------
// Kernel 5: output layer: logits -> softmax -> probs; then (as the reference
// does) log_softmax of the *probs* and summed NLL. One block of 256 threads.
// d_out layout: probs[G*2] then loss (1 float) => 513 floats.
// ---------------------------------------------------------------------------
__global__ void head_kernel(const float* __restrict__ h,
                            const int* __restrict__ slot_of,
                            const int* __restrict__ label,
                            const float* __restrict__ W_out,
                            const float* __restrict__ b_out,
                            float* __restrict__ out) {
    __shared__ float red[GG];
    int g = threadIdx.x;
    int slot = slot_of[g];
    const float* hr = h + slot * HDIM;

    float l0 = b_out[0], l1 = b_out[1];
    for (int i = 0; i < HDIM; ++i) {
        float v = hr[i];
        l0 += v * W_out[i * 2 + 0];
        l1 += v * W_out[i * 2 + 1];
    }
    float mx = fmaxf(l0, l1);
    float e0 = expf(l0 - mx), e1 = expf(l1 - mx);
    float inv = 1.0f / (e0 + e1);
    float p0 = e0 * inv, p1 = e1 * inv;
    out[2 * g + 0] = p0;
    out[2 * g + 1] = p1;

    // log_softmax applied to probs (faithful to the reference)
    float mm = fmaxf(p0, p1);
    float q0 = p0 - mm, q1 = p1 - mm;
    float lse = logf(expf(q0) + expf(q1));
    float lp = (label[g] == 0 ? q0 : q1) - lse;
    red[g] = -lp;
    __syncthreads();
    for (int s = GG / 2; s > 0; s >>= 1) {
        if (g < s) red[g] += red[g + s];
        __syncthreads();
    }
    if (g == 0) out[2 * GG] = red[0];
}

// ---------------------------------------------------------------------------
extern "C" void kernel_launch(void* const* d_in, const int* in_sizes, int n_in,
                              void* d_out, int out_size, void* d_ws, size_t ws_size,
                              hipStream_t stream) {
    const float* node_fea = (const float*)d_in[0];
    const float* masks    = (const float*)d_in[1];
    const int*   node_type= (const int*)  d_in[2];
    const int*   g_node_id= (const int*)  d_in[3];
    const int*   label    = (const int*)  d_in[4];
    const float* W_emb    = (const float*)d_in[5];
    const float* b_emb    = (const float*)d_in[6];
    const float* W_hid    = (const float*)d_in[7];
    const float* b_hid    = (const float*)d_in[8];
    const float* W_out    = (const float*)d_in[9];
    const float* b_out    = (const float*)d_in[10];
    (void)in_sizes; (void)n_in; (void)out_size; (void)ws_size;

    char* ws = (char*)d_ws;
    int*   perm    = (int*)  (ws + WS_PERM);
    int*   slot_of = (int*)  (ws + WS_SLOTOF);
    int*   eoff    = (int*)  (ws + WS_OFF);
    float* xg      = (float*)(ws + WS_XG);
    float* emb     = (float*)(ws + WS_EMB);
    float* h       = (float*)(ws + WS_H);
    float* out     = (float*)d_out;

    bucket_kernel<<<1, GG, 0, stream>>>(node_type, g_node_id, perm, slot_of, eoff);
    gather_kernel<<<GG + PADR, FDIM, 0, stream>>>(node_fea, masks, g_node_id, perm, xg);
    expert_gemm_kernel<<<dim3(DDIM / 64, GG / 16, EXP), 32, 0, stream>>>(
        xg, W_emb, b_emb, eoff, emb);
    hidden_gemm_kernel<<<dim3(HDIM / 64, GG / 16), 32, 0, stream>>>(
        emb, W_hid, b_hid, h);
    head_kernel<<<1, GG, 0, stream>>>(h, slot_of, label, W_out, b_out, out);
}